// RadixAttention_28595892257092
// MI455X (gfx1250) — compile-verified
//
#include <hip/hip_runtime.h>

#define B_    32
#define S_    2048
#define H_    32
#define HKV_  8
#define G_    4
#define D_    128
#define NW    8          // waves per block (wave32)
#define WSZ   32
#define NEGV  (-1.0e30f)
// 1/sqrt(128) * log2(e): fold log2e in so softmax uses raw v_exp_f32 (exp2)
#define QSCALE (0.08838834764831845f * 1.4426950408889634f)

typedef __attribute__((ext_vector_type(2))) float v2f;
typedef __attribute__((ext_vector_type(8))) float v8f;

__global__ __launch_bounds__(NW * WSZ)
void radix_decode_attn(const float* __restrict__ q,
                       const float* __restrict__ knew,
                       const float* __restrict__ vnew,
                       const float* __restrict__ kbuf,
                       const float* __restrict__ vbuf,
                       const int*   __restrict__ r2t,
                       const int*   __restrict__ seqlens,
                       const int*   __restrict__ newloc,
                       float*       __restrict__ out)
{
    __shared__ float o_lds[G_ * D_];        // merged output accumulator
    __shared__ float ml_lds[NW * G_ * 2];   // per-wave (m, l) per head
    __shared__ float L_lds[G_];             // merged denominator

    const int blk  = blockIdx.x;
    const int b    = blk / HKV_;
    const int kvh  = blk % HKV_;
    const int tid  = threadIdx.x;
    const int w    = tid >> 5;      // wave id
    const int lane = tid & 31;
    const int lo   = lane & 15;     // M/N index within half
    const int hi   = lane >> 4;     // lane-half

    // zero shared output accumulator (first __syncthreads covers visibility)
    for (int i = tid; i < G_ * D_; i += NW * WSZ) o_lds[i] = 0.0f;

    const int  n    = seqlens[b];
    const int  ntok = newloc[b];
    const long rstr = (long)HKV_ * D_;                       // KV pool row stride
    const float* krow_new = knew + ((long)b * HKV_ + kvh) * D_;
    const float* vrow_new = vnew + ((long)b * HKV_ + kvh) * D_;
    const float* kb = kbuf + (long)kvh * D_;
    const float* vb = vbuf + (long)kvh * D_;
    const int*   tok_b = r2t + b * S_;

    // ---- Q in WMMA B-operand layout, persistent in VGPRs -------------------
    // vb0: low lanes K=d0, high lanes K=d0+2 ; vb1: d0+1 / d0+3  (N = head = lo)
    v2f qb[32];
    {
        const float* qp = q + ((long)b * H_ + kvh * G_ + lo) * D_;
        #pragma unroll
        for (int c = 0; c < 32; ++c) {
            v2f t = {0.0f, 0.0f};
            if (lo < G_) t = *(const v2f*)(qp + 4 * c + 2 * hi);
            qb[c].x = t.x * QSCALE;
            qb[c].y = t.y * QSCALE;
        }
    }

    // ---- per-wave sequence range (multiple of 16) --------------------------
    const int tiles  = (n + 16 * NW - 1) / (16 * NW);
    const int wstart = w * tiles * 16;
    const int wend   = min(n, wstart + tiles * 16);

    v8f oacc[8];                 // O^T accumulator: (M = d-in-chunk, N = head)
    #pragma unroll
    for (int i = 0; i < 8; ++i) oacc[i] = v8f{};
    float m_run = -3.0e38f;
    float lsum  = 0.0f;

    for (int s0 = wstart; s0 < wend; s0 += 16) {
        // K row pointer for this lane's position row (M = lo)
        const int tk = tok_b[s0 + lo];
        const float* krow = (tk == ntok) ? krow_new : (kb + (long)tk * rstr);
        const float* kptr = krow + 2 * hi;

        // ---- scores(pos,head) = K_tile(16x128) . Q^T(128x16) ----
        // K rows are single-use, 537MB stream >> 192MB L2 -> non-temporal
        v8f sc = v8f{};
        #pragma unroll
        for (int c = 0; c < 32; ++c) {
            v2f ka = __builtin_nontemporal_load((const v2f*)(kptr + 4 * c));
            sc = __builtin_amdgcn_wmma_f32_16x16x4_f32(
                     false, ka, false, qb[c], (short)0, sc, false, false);
        }

        // mask tail positions (wave-uniform branch; only last tile pays)
        if (s0 + 16 > wend) {
            #pragma unroll
            for (int r = 0; r < 8; ++r) {
                const int pos = s0 + r + 8 * hi;    // C layout: M = r + 8*hi
                if (pos >= wend) sc[r] = NEGV;
            }
        }

        // ---- online softmax (per head = per lane column, halves via xor16) --
        float mt = sc[0];
        #pragma unroll
        for (int r = 1; r < 8; ++r) mt = fmaxf(mt, sc[r]);
        mt = fmaxf(mt, __shfl_xor(mt, 16, WSZ));
        const float mnew = fmaxf(m_run, mt);
        // args are <= 0: raw v_exp_f32 (no libm range fixup needed)
        const float corr = __builtin_amdgcn_exp2f(m_run - mnew);
        m_run = mnew;
        float psum = 0.0f;
        #pragma unroll
        for (int r = 0; r < 8; ++r) {
            const float p = __builtin_amdgcn_exp2f(sc[r] - mnew); // masked -> 0
            sc[r] = p;
            psum += p;
        }
        lsum = lsum * corr + psum;
        #pragma unroll
        for (int i = 0; i < 8; ++i)
            #pragma unroll
            for (int r = 0; r < 8; ++r) oacc[i][r] *= corr;

        // ---- rebuild P into B-operand layout (K = position chunks of 4) ----
        float sw[8];
        #pragma unroll
        for (int r = 0; r < 8; ++r) sw[r] = __shfl_xor(sc[r], 16, WSZ);
        v2f pb[4];
        pb[0].x = hi ? sw[2] : sc[0];  pb[0].y = hi ? sw[3] : sc[1];
        pb[1].x = hi ? sw[6] : sc[4];  pb[1].y = hi ? sw[7] : sc[5];
        pb[2].x = hi ? sc[2] : sw[0];  pb[2].y = hi ? sc[3] : sw[1];
        pb[3].x = hi ? sc[6] : sw[4];  pb[3].y = hi ? sc[7] : sw[5];

        // ---- O^T += V^T_tile . P : A = V^T (M=d, K=pos), 32 WMMAs ----------
        #pragma unroll
        for (int c = 0; c < 4; ++c) {
            const int p0 = 4 * c + 2 * hi;          // A layout: K pair per half
            const int t0 = tok_b[s0 + p0];
            const int t1 = tok_b[s0 + p0 + 1];
            const float* vr0 = ((t0 == ntok) ? vrow_new : (vb + (long)t0 * rstr)) + lo;
            const float* vr1 = ((t1 == ntok) ? vrow_new : (vb + (long)t1 * rstr)) + lo;
            #pragma unroll
            for (int dc = 0; dc < 8; ++dc) {
                v2f va;
                va.x = __builtin_nontemporal_load(vr0 + dc * 16); // coalesced b32
                va.y = __builtin_nontemporal_load(vr1 + dc * 16);
                oacc[dc] = __builtin_amdgcn_wmma_f32_16x16x4_f32(
                               false, va, false, pb[c], (short)0, oacc[dc],
                               false, false);
            }
        }
    }

    // ---- cross-wave log-sum-exp merge via LDS ------------------------------
    lsum += __shfl_xor(lsum, 16, WSZ);              // full per-head denominator
    if (lane < G_) {
        ml_lds[(w * G_ + lane) * 2 + 0] = m_run;
        ml_lds[(w * G_ + lane) * 2 + 1] = lsum;
    }
    __syncthreads();

    float factor = 0.0f;
    if (lo < G_) {
        float mf = -3.0e38f;
        #pragma unroll
        for (int ww = 0; ww < NW; ++ww)
            mf = fmaxf(mf, ml_lds[(ww * G_ + lo) * 2]);
        float Lt = 0.0f;
        #pragma unroll
        for (int ww = 0; ww < NW; ++ww)
            Lt += ml_lds[(ww * G_ + lo) * 2 + 1] *
                  __builtin_amdgcn_exp2f(ml_lds[(ww * G_ + lo) * 2] - mf);
        factor = __builtin_amdgcn_exp2f(m_run - mf);
        if (w == 0 && lane < G_) L_lds[lane] = Lt;
        // accumulate this wave's O into shared (ds_add_f32 atomics)
        #pragma unroll
        for (int dc = 0; dc < 8; ++dc)
            #pragma unroll
            for (int r = 0; r < 8; ++r)
                atomicAdd(&o_lds[lo * D_ + dc * 16 + r + 8 * hi],
                          oacc[dc][r] * factor);
    }
    __syncthreads();

    // ---- normalize + store: out[b][(kvh*G + h)*D + d] ----------------------
    float* outp = out + ((long)b * H_ + kvh * G_) * D_;
    for (int i = tid; i < G_ * D_; i += NW * WSZ) {
        const int h = i / D_;
        outp[i] = o_lds[i] / L_lds[h];
    }
}

extern "C" void kernel_launch(void* const* d_in, const int* in_sizes, int n_in,
                              void* d_out, int out_size, void* d_ws, size_t ws_size,
                              hipStream_t stream) {
    const float* q    = (const float*)d_in[0];
    const float* k    = (const float*)d_in[1];
    const float* v    = (const float*)d_in[2];
    const float* kbuf = (const float*)d_in[3];
    const float* vbuf = (const float*)d_in[4];
    const int*   r2t  = (const int*)d_in[5];
    const int*   sl   = (const int*)d_in[6];
    const int*   ocl  = (const int*)d_in[7];
    float*       out  = (float*)d_out;

    dim3 grid(B_ * HKV_);
    dim3 block(NW * WSZ);
    hipLaunchKernelGGL(radix_decode_attn, grid, block, 0, stream,
                       q, k, v, kbuf, vbuf, r2t, sl, ocl, out);
}